// GAT_25907242729853
// MI455X (gfx1250) — compile-verified
//
#include <hip/hip_runtime.h>
#include <math.h>

// Problem constants (from reference)
#define NN      50000
#define EE      800000
#define INDIM   128
#define ODIM    64
#define H1      2       // heads layer 1
#define NEGS    0.2f

typedef float v2f __attribute__((ext_vector_type(2)));
typedef float v8f __attribute__((ext_vector_type(8)));

// -------------------------------------------------------------------------
// WMMA fp32 GEMM:  Hout[n x Mo] = X[n x K] @ W[K x Mo]
// One wave computes one 16x16 output tile with V_WMMA_F32_16X16X4_F32.
// Requires n%16==0 (50000=3125*16), Mo%16==0, K%4==0. Guard is wave-uniform
// so EXEC is all-ones at every WMMA (ISA requirement).
// -------------------------------------------------------------------------
__global__ __launch_bounds__(256) void gemm_wmma_f32(
    const float* __restrict__ X, const float* __restrict__ W,
    float* __restrict__ Hout, int n, int K, int Mo)
{
    const int lane = threadIdx.x & 31;
    const int wave = threadIdx.x >> 5;
    const int colTiles = Mo >> 4;
    const int numTiles = (n >> 4) * colTiles;
    const int tile = blockIdx.x * 8 + wave;
    if (tile >= numTiles) return;                       // wave-uniform exit

    const int rowBase = (tile / colTiles) << 4;
    const int colBase = (tile % colTiles) << 4;
    const int m  = lane & 15;
    const int ko = (lane < 16) ? 0 : 2;                 // K-offset per half-wave

    const float* __restrict__ xrow = X + (size_t)(rowBase + m) * K + ko;
    const float* __restrict__ wcol = W + (size_t)ko * Mo + colBase + m;

    v8f c = {0.f, 0.f, 0.f, 0.f, 0.f, 0.f, 0.f, 0.f};
    for (int k = 0; k < K; k += 4) {
        // A 16x4 fragment: VGPR v holds K = ko+v for row M = lane%16
        v2f a;
        a.x = xrow[k];
        a.y = xrow[k + 1];
        // B 4x16 fragment: VGPR v holds K = ko+v, N = lane%16
        v2f b;
        b.x = wcol[(size_t)k * Mo];
        b.y = wcol[(size_t)(k + 1) * Mo];
        c = __builtin_amdgcn_wmma_f32_16x16x4_f32(
                /*neg_a=*/false, a, /*neg_b=*/false, b,
                /*c_mod=*/(short)0, c, /*reuse_a=*/false, /*reuse_b=*/false);
    }

    // D layout: VGPR r -> M=r (lanes 0-15) / M=8+r (lanes 16-31); N = lane%16
    const int rOff = (lane < 16) ? 0 : 8;
    float* __restrict__ orow = Hout + colBase + m;
#pragma unroll
    for (int r = 0; r < 8; ++r)
        orow[(size_t)(rowBase + rOff + r) * Mo] = c[r];
}

// -------------------------------------------------------------------------
// el[i] = <h[n,h,:], al[h,:]>, er likewise. i runs over n*H.
// -------------------------------------------------------------------------
__global__ void attn_dots(const float* __restrict__ Hf,
                          const float* __restrict__ al,
                          const float* __restrict__ ar,
                          float* __restrict__ el, float* __restrict__ er,
                          int n, int H)
{
    int i = blockIdx.x * blockDim.x + threadIdx.x;
    if (i >= n * H) return;
    int h = i % H;
    const float* hv = Hf + (size_t)(i / H) * (H * ODIM) + h * ODIM;
    const float* a1 = al + h * ODIM;
    const float* a2 = ar + h * ODIM;
    float s1 = 0.f, s2 = 0.f;
#pragma unroll 4
    for (int d = 0; d < ODIM; ++d) {
        float v = hv[d];
        s1 += v * a1[d];
        s2 += v * a2[d];
    }
    el[i] = s1;
    er[i] = s2;
}

__global__ void fill_f32(float* __restrict__ p, float v, int n)
{
    int i = blockIdx.x * blockDim.x + threadIdx.x;
    if (i < n) p[i] = v;
}
__global__ void fill_u32(unsigned* __restrict__ p, unsigned v, int n)
{
    int i = blockIdx.x * blockDim.x + threadIdx.x;
    if (i < n) p[i] = v;
}

// Order-preserving float <-> uint key for atomic max of signed floats.
__device__ __forceinline__ unsigned fkey(float f)
{
    unsigned b = __float_as_uint(f);
    return b ^ ((b >> 31) ? 0xFFFFFFFFu : 0x80000000u);
}
__device__ __forceinline__ float funkey(unsigned k)
{
    unsigned b = k ^ ((k >> 31) ? 0x80000000u : 0xFFFFFFFFu);
    return __uint_as_float(b);
}
#define MINKEY 0x007FFFFFu   // fkey(-inf)

// Pass 1: e = leaky_relu(el[src]+er[dst]); store e; segment max by dst.
__global__ void edge_max(const float* __restrict__ el, const float* __restrict__ er,
                         const int* __restrict__ src, const int* __restrict__ dst,
                         float* __restrict__ ebuf, unsigned* __restrict__ mkey,
                         int E, int H)
{
    int i = blockIdx.x * blockDim.x + threadIdx.x;
    if (i >= E * H) return;
    int e = i / H, h = i % H;
    float t = el[src[e] * H + h] + er[dst[e] * H + h];
    t = (t > 0.f) ? t : NEGS * t;
    ebuf[i] = t;
    atomicMax(&mkey[dst[e] * H + h], fkey(t));
}

// Decode segment max in place; guard isolated nodes (-inf -> 0), as reference.
__global__ void finalize_m(unsigned* __restrict__ mk, float* __restrict__ mf, int n)
{
    int i = blockIdx.x * blockDim.x + threadIdx.x;
    if (i >= n) return;
    float m = funkey(mk[i]);
    mf[i] = (m >= -3.0e38f) ? m : 0.f;
}

// Pass 2: ex = exp(e - m[dst]); store ex; segment sum by dst.
__global__ void edge_expsum(const int* __restrict__ dst,
                            const float* __restrict__ m,
                            float* __restrict__ ebuf, float* __restrict__ s,
                            int E, int H)
{
    int i = blockIdx.x * blockDim.x + threadIdx.x;
    if (i >= E * H) return;
    int e = i / H, h = i % H;
    float ex = __expf(ebuf[i] - m[dst[e] * H + h]);
    ebuf[i] = ex;
    atomicAdd(&s[dst[e] * H + h], ex);
}

// Pass 3: one wave per (edge, head); 2 feature dims per lane; fp32 atomics.
__global__ __launch_bounds__(256) void edge_scatter(
    const int* __restrict__ src, const int* __restrict__ dst,
    const float* __restrict__ ebuf, const float* __restrict__ s,
    const float* __restrict__ Hf, float* __restrict__ acc, int E, int H)
{
    int lane = threadIdx.x & 31;
    int w = (blockIdx.x * blockDim.x + threadIdx.x) >> 5;
    if (w >= E * H) return;
    int e = w / H, h = w % H;
    int sN = src[e], dN = dst[e];
    float alpha = ebuf[w] / s[dN * H + h];
    const float* hs = Hf + (size_t)sN * (H * ODIM) + h * ODIM;
    float* ad = acc + (size_t)dN * (H * ODIM) + h * ODIM;
    atomicAdd(&ad[lane],      alpha * hs[lane]);
    atomicAdd(&ad[lane + 32], alpha * hs[lane + 32]);
}

// out = elu(acc + bias)  (layer-1 epilogue; F = H*ODIM)
__global__ void bias_elu(const float* __restrict__ acc, const float* __restrict__ b,
                         float* __restrict__ out, int n, int F)
{
    int i = blockIdx.x * blockDim.x + threadIdx.x;
    if (i >= n * F) return;
    float v = acc[i] + b[i % F];
    out[i] = (v > 0.f) ? v : expm1f(v);
}

// final: out = (mean(h1 heads) + (acc2 + b2)) * 0.5
__global__ void final_combine(const float* __restrict__ h1,
                              const float* __restrict__ acc2,
                              const float* __restrict__ b2,
                              float* __restrict__ out, int n)
{
    int i = blockIdx.x * blockDim.x + threadIdx.x;
    if (i >= n * ODIM) return;
    int nn = i / ODIM, d = i % ODIM;
    float mean1 = 0.5f * (h1[(size_t)nn * INDIM + d] + h1[(size_t)nn * INDIM + ODIM + d]);
    float h2v = acc2[i] + b2[d];
    out[i] = (mean1 + h2v) * 0.5f;
}

static inline int cdiv(long long a, long long b) { return (int)((a + b - 1) / b); }

extern "C" void kernel_launch(void* const* d_in, const int* in_sizes, int n_in,
                              void* d_out, int out_size, void* d_ws, size_t ws_size,
                              hipStream_t stream)
{
    const float* feature = (const float*)d_in[0];
    const int*   src     = (const int*)  d_in[1];
    const int*   dst     = (const int*)  d_in[2];
    const float* W1      = (const float*)d_in[3];
    const float* al1     = (const float*)d_in[4];
    const float* ar1     = (const float*)d_in[5];
    const float* b1      = (const float*)d_in[6];
    const float* W2      = (const float*)d_in[7];
    const float* al2     = (const float*)d_in[8];
    const float* ar2     = (const float*)d_in[9];
    const float* b2      = (const float*)d_in[10];
    float* out = (float*)d_out;

    // Workspace layout (floats)
    float* ws = (float*)d_ws;
    size_t o = 0;
    float* h1   = ws + o; o += (size_t)NN * INDIM;   // GEMM1 out, later elu'd h1
    float* acc1 = ws + o; o += (size_t)NN * INDIM;   // layer-1 scatter accumulator
    float* h2   = ws + o; o += (size_t)NN * ODIM;    // GEMM2 out
    float* acc2 = ws + o; o += (size_t)NN * ODIM;    // layer-2 scatter accumulator
    float* el1  = ws + o; o += (size_t)NN * H1;
    float* er1  = ws + o; o += (size_t)NN * H1;
    float* m1   = ws + o; o += (size_t)NN * H1;      // mkey (u32) then decoded m
    float* s1   = ws + o; o += (size_t)NN * H1;
    float* el2  = ws + o; o += (size_t)NN;
    float* er2  = ws + o; o += (size_t)NN;
    float* m2   = ws + o; o += (size_t)NN;
    float* s2   = ws + o; o += (size_t)NN;
    float* e1   = ws + o; o += (size_t)EE * H1;      // per-edge e / ex, layer 1
    float* e2   = ws + o; o += (size_t)EE;           // per-edge e / ex, layer 2

    const int T = 256;

    // ---------------- Layer 1 (H=2) ----------------
    // GEMM: feature[N,128] @ W1[128,128] -> h1
    {
        int tiles = (NN / 16) * (INDIM / 16);
        gemm_wmma_f32<<<cdiv(tiles, 8), T, 0, stream>>>(feature, W1, h1, NN, INDIM, INDIM);
    }
    attn_dots<<<cdiv((long long)NN * H1, T), T, 0, stream>>>(h1, al1, ar1, el1, er1, NN, H1);

    fill_f32<<<cdiv((long long)NN * INDIM, T), T, 0, stream>>>(acc1, 0.f, NN * INDIM);
    fill_f32<<<cdiv((long long)NN * H1, T), T, 0, stream>>>(s1, 0.f, NN * H1);
    fill_u32<<<cdiv((long long)NN * H1, T), T, 0, stream>>>((unsigned*)m1, MINKEY, NN * H1);

    edge_max<<<cdiv((long long)EE * H1, T), T, 0, stream>>>(el1, er1, src, dst,
                                                            e1, (unsigned*)m1, EE, H1);
    finalize_m<<<cdiv((long long)NN * H1, T), T, 0, stream>>>((unsigned*)m1, m1, NN * H1);
    edge_expsum<<<cdiv((long long)EE * H1, T), T, 0, stream>>>(dst, m1, e1, s1, EE, H1);
    edge_scatter<<<cdiv((long long)EE * H1 * 32, T), T, 0, stream>>>(src, dst, e1, s1,
                                                                     h1, acc1, EE, H1);
    // h1 <- elu(acc1 + b1) (overwrites GEMM1 output; safe, scatter is done)
    bias_elu<<<cdiv((long long)NN * INDIM, T), T, 0, stream>>>(acc1, b1, h1, NN, INDIM);

    // ---------------- Layer 2 (H=1) ----------------
    {
        int tiles = (NN / 16) * (ODIM / 16);
        gemm_wmma_f32<<<cdiv(tiles, 8), T, 0, stream>>>(h1, W2, h2, NN, INDIM, ODIM);
    }
    attn_dots<<<cdiv((long long)NN, T), T, 0, stream>>>(h2, al2, ar2, el2, er2, NN, 1);

    fill_f32<<<cdiv((long long)NN * ODIM, T), T, 0, stream>>>(acc2, 0.f, NN * ODIM);
    fill_f32<<<cdiv((long long)NN, T), T, 0, stream>>>(s2, 0.f, NN);
    fill_u32<<<cdiv((long long)NN, T), T, 0, stream>>>((unsigned*)m2, MINKEY, NN);

    edge_max<<<cdiv((long long)EE, T), T, 0, stream>>>(el2, er2, src, dst,
                                                       e2, (unsigned*)m2, EE, 1);
    finalize_m<<<cdiv((long long)NN, T), T, 0, stream>>>((unsigned*)m2, m2, NN);
    edge_expsum<<<cdiv((long long)EE, T), T, 0, stream>>>(dst, m2, e2, s2, EE, 1);
    edge_scatter<<<cdiv((long long)EE * 32, T), T, 0, stream>>>(src, dst, e2, s2,
                                                                h2, acc2, EE, 1);

    final_combine<<<cdiv((long long)NN * ODIM, T), T, 0, stream>>>(h1, acc2, b2, out, NN);
}